// WeaveGather_33904471834983
// MI455X (gfx1250) — compile-verified
//
#include <hip/hip_runtime.h>

#define N_ATOMS 131072
#define N_INPUT 128
#define N_BINS  11
#define HDIM    (N_INPUT * N_BINS)   // 1408
#define BATCH   4096

typedef __attribute__((ext_vector_type(2))) float v2f;
typedef __attribute__((ext_vector_type(8))) float v8f;

// -----------------------------------------------------------------------------
// Kernel 1: fused gaussian-histogram + segment-sum.
// One block per molecule (segment). membership is sorted, so the segment is a
// contiguous atom range found by binary search. Thread f (0..127) owns feature
// f and keeps 11 accumulators in registers; writes mol[seg][f*11 + j].
// Every row of mol is fully overwritten each call (deterministic, no zeroing).
// -----------------------------------------------------------------------------
__global__ __launch_bounds__(128) void weave_hist_segsum(
    const float* __restrict__ atom_features,
    const int*   __restrict__ membership,
    float*       __restrict__ mol) {
  const float MU[N_BINS] = {-1.645f, -1.08f, -0.739f, -0.468f, -0.228f, 0.0f,
                             0.228f,  0.468f, 0.739f,  1.08f,  1.645f};
  const float SIG[N_BINS] = {0.283f, 0.17f, 0.134f, 0.118f, 0.114f, 0.114f,
                             0.114f, 0.118f, 0.134f, 0.17f, 0.283f};

  const int seg = blockIdx.x;

  // lower_bound(membership, seg)
  int lo = 0, hi = N_ATOMS;
  while (lo < hi) {
    int mid = (lo + hi) >> 1;
    if (membership[mid] < seg) lo = mid + 1; else hi = mid;
  }
  const int start = lo;
  // upper_bound(membership, seg)
  hi = N_ATOMS;
  while (lo < hi) {
    int mid = (lo + hi) >> 1;
    if (membership[mid] <= seg) lo = mid + 1; else hi = mid;
  }
  const int end = lo;

  const int f = threadIdx.x;  // feature index 0..127

  float acc[N_BINS];
#pragma unroll
  for (int j = 0; j < N_BINS; ++j) acc[j] = 0.0f;

  for (int a = start; a < end; ++a) {
    const float x = atom_features[(size_t)a * N_INPUT + f];
    float g[N_BINS];
    float s = 0.0f;
#pragma unroll
    for (int j = 0; j < N_BINS; ++j) {
      const float d = (x - MU[j]) / SIG[j];           // folded to mul by const
      g[j] = __expf(-0.5f * d * d);                   // v_exp_f32 path
      s += g[j];
    }
    const float inv = __builtin_amdgcn_rcpf(s);       // v_rcp_f32
#pragma unroll
    for (int j = 0; j < N_BINS; ++j) acc[j] = fmaf(g[j], inv, acc[j]);
  }

  float* dst = mol + (size_t)seg * HDIM + (size_t)f * N_BINS;
#pragma unroll
  for (int j = 0; j < N_BINS; ++j) dst[j] = acc[j];
}

// -----------------------------------------------------------------------------
// Kernel 2: out = tanh(mol @ W + b) via V_WMMA_F32_16X16X4_F32 (full f32 MAC).
// Block = 256 threads = 8 wave32s. Block b covers output rows [16b,16b+16);
// wave w covers output cols [16w,16w+16). K loop steps by 4 (352 iters).
//
// ISA 7.12.2 f32 layouts (wave32):
//   A 16x4:  lanes 0-15 -> M=lane, {v0,v1}={K0,K1}; lanes 16-31 -> {K2,K3}
//   B 4x16:  lanes 0-15 -> N=lane, {v0,v1}={K0,K1}; lanes 16-31 -> {K2,K3}
//   C/D 16x16: v[r], lanes 0-15 -> M=r, N=lane; lanes 16-31 -> M=r+8
// -----------------------------------------------------------------------------
__global__ __launch_bounds__(256) void weave_gemm_tanh(
    const float* __restrict__ mol,
    const float* __restrict__ W,
    const float* __restrict__ bias,
    float*       __restrict__ out) {
  const int rowBase = blockIdx.x << 4;
  const int lane    = threadIdx.x & 31;
  const int wave    = threadIdx.x >> 5;      // 0..7
  const int colBase = wave << 4;
  const int half    = lane >> 4;             // 0: K0/K1 group, 1: K2/K3 group
  const int l       = lane & 15;             // M (for A) / N (for B) index

  // A: row (rowBase+l), columns k + 2*half + {0,1}  -> aligned float2
  const float* Arow = mol + (size_t)(rowBase + l) * HDIM + (half << 1);
  // B: rows k + 2*half + {0,1}, column colBase+l
  const float* Bcol = W + (size_t)(half << 1) * N_INPUT + colBase + l;

  v8f c = {0.f, 0.f, 0.f, 0.f, 0.f, 0.f, 0.f, 0.f};

#pragma unroll 4
  for (int k = 0; k < HDIM; k += 4) {
    const v2f a = *(const v2f*)(Arow + k);
    v2f bf;
    bf.x = Bcol[(size_t)k * N_INPUT];
    bf.y = Bcol[(size_t)(k + 1) * N_INPUT];
    // 8 args: (neg_a, A, neg_b, B, c_mod, C, reuse_a, reuse_b)
    c = __builtin_amdgcn_wmma_f32_16x16x4_f32(false, a, false, bf,
                                              (short)0, c, false, false);
  }

  const float bv = bias[colBase + l];
#pragma unroll
  for (int r = 0; r < 8; ++r) {
    const int row = rowBase + r + (half << 3);
    out[(size_t)row * N_INPUT + colBase + l] = tanhf(c[r] + bv);
  }
}

// -----------------------------------------------------------------------------
// Launch. Inputs: atom_features f32[131072*128], membership i32[131072],
// W f32[1408*128], b f32[128], batch_size (ignored scalar).
// Workspace: mol f32[4096*1408] = 23.1 MB.
// -----------------------------------------------------------------------------
extern "C" void kernel_launch(void* const* d_in, const int* in_sizes, int n_in,
                              void* d_out, int out_size, void* d_ws, size_t ws_size,
                              hipStream_t stream) {
  const float* atom_features = (const float*)d_in[0];
  const int*   membership    = (const int*)d_in[1];
  const float* W             = (const float*)d_in[2];
  const float* b             = (const float*)d_in[3];
  float*       out           = (float*)d_out;
  float*       mol           = (float*)d_ws;

  weave_hist_segsum<<<BATCH, 128, 0, stream>>>(atom_features, membership, mol);
  weave_gemm_tanh<<<BATCH / 16, 256, 0, stream>>>(mol, W, b, out);
}